// LsqNonneg_15040975470881
// MI455X (gfx1250) — compile-verified
//
#include <hip/hip_runtime.h>
#include <math.h>

#define M_DIM 1024
#define K_DIM 32
#define N_DIM 16384
#define FISTA_ITERS 500

typedef __attribute__((ext_vector_type(2))) float v2f;
typedef __attribute__((ext_vector_type(8))) float v8f;

// Workspace layout (floats):
//   [0, 1024)            : G (32x32, row-major)
//   [1024]               : step = 1/L
//   [2048, 2048+32*16384): AtX (32 x 16384, row-major)

// ---------------------------------------------------------------------------
// Kernel 1: G = A^T A  +  power iteration  ->  step = 1/L
// One workgroup of 1024 threads; wave w computes row w of G (lane = column),
// A[m*32+j] is a coalesced 128B line per wave, A[m*32+i] is a wave-uniform
// broadcast. Power iteration done by the first 32 threads through LDS.
// ---------------------------------------------------------------------------
__global__ __launch_bounds__(1024) void gram_step_kernel(
    const float* __restrict__ A, float* __restrict__ Gout,
    float* __restrict__ stepOut) {
  __shared__ float Gs[K_DIM * K_DIM];
  __shared__ float vs[K_DIM];
  __shared__ float wsv[K_DIM];
  const int t = threadIdx.x;
  const int i = t >> 5;
  const int j = t & 31;
  float acc = 0.f;
  for (int m = 0; m < M_DIM; ++m)
    acc = fmaf(A[m * K_DIM + i], A[m * K_DIM + j], acc);
  Gs[i * K_DIM + j] = acc;
  Gout[i * K_DIM + j] = acc;
  __syncthreads();
  if (t < K_DIM) vs[t] = 0.17677669529663687f;  // 1/sqrt(32)
  __syncthreads();
  for (int it = 0; it < 100; ++it) {
    if (t < K_DIM) {
      float w = 0.f;
      #pragma unroll
      for (int c = 0; c < K_DIM; ++c) w = fmaf(Gs[t * K_DIM + c], vs[c], w);
      wsv[t] = w;
    }
    __syncthreads();
    if (t < K_DIM) {
      float nrm = 0.f;
      #pragma unroll
      for (int c = 0; c < K_DIM; ++c) nrm = fmaf(wsv[c], wsv[c], nrm);
      nrm = sqrtf(nrm) + 1e-30f;
      vs[t] = wsv[t] / nrm;
    }
    __syncthreads();
  }
  if (t == 0) {
    float L = 0.f;
    for (int r = 0; r < K_DIM; ++r) {
      float gv = 0.f;
      #pragma unroll
      for (int c = 0; c < K_DIM; ++c) gv = fmaf(Gs[r * K_DIM + c], vs[c], gv);
      L = fmaf(vs[r], gv, L);
    }
    stepOut[0] = 1.0f / L;
  }
}

// ---------------------------------------------------------------------------
// Kernel 2: AtX = A^T X   (bandwidth-bound: streams X exactly once, 64 MB)
// Thread owns one column n; A staged through LDS in 128-row chunks.
// X read with a non-temporal hint (read-once stream).
// ---------------------------------------------------------------------------
__global__ __launch_bounds__(256) void atx_kernel(const float* __restrict__ X,
                                                  const float* __restrict__ A,
                                                  float* __restrict__ AtX) {
  __shared__ float As[128 * K_DIM];  // 16 KB
  const int n = blockIdx.x * 256 + threadIdx.x;
  float acc[K_DIM];
  #pragma unroll
  for (int k = 0; k < K_DIM; ++k) acc[k] = 0.f;
  for (int chunk = 0; chunk < M_DIM / 128; ++chunk) {
    #pragma unroll
    for (int i = 0; i < 16; ++i) {
      const int idx = threadIdx.x + i * 256;
      As[idx] = A[chunk * (128 * K_DIM) + idx];
    }
    __syncthreads();
    for (int mm = 0; mm < 128; ++mm) {
      const float x = __builtin_nontemporal_load(&X[(chunk * 128 + mm) * N_DIM + n]);
      #pragma unroll
      for (int k = 0; k < K_DIM; ++k)
        acc[k] = fmaf(As[mm * K_DIM + k], x, acc[k]);
    }
    __syncthreads();
  }
  #pragma unroll
  for (int k = 0; k < K_DIM; ++k) AtX[k * N_DIM + n] = acc[k];
}

// ---------------------------------------------------------------------------
// Kernel 3: FISTA loop, fully register/LDS-resident per wave.
// Each wave owns 16 columns. G kept in WMMA A-operand layout (constant
// across iterations); Y/S/AtX kept in C/D accumulator layout; per iteration
// Y is staged through a double-buffered per-wave LDS tile to regenerate the
// B-operand (4x16) layout, then G@Y = 16 x v_wmma_f32_16x16x4_f32.
// Double buffering removes the anti-dependency barrier: one barrier/iter.
// ---------------------------------------------------------------------------
__global__ __launch_bounds__(128) void fista_kernel(
    const float* __restrict__ Gws, const float* __restrict__ stepWs,
    const float* __restrict__ AtX, float* __restrict__ Sout) {
  __shared__ float Ybuf[2][4][K_DIM * 16];  // 2 x per-wave 2 KB staging tile
  const int lane = threadIdx.x & 31;
  const int wave = threadIdx.x >> 5;
  const int colLoc = lane & 15;
  const int half = lane >> 4;
  const int col = blockIdx.x * 64 + wave * 16 + colLoc;
  const float step = stepWs[0];

  // G in A-operand layout: 16x4 tile, VGPR j: lanes 0-15 K=4c+j, lanes 16-31 K=4c+2+j
  v2f ga[2][8];
  #pragma unroll
  for (int r = 0; r < 2; ++r) {
    #pragma unroll
    for (int c = 0; c < 8; ++c) {
      const int row = r * 16 + colLoc;
      #pragma unroll
      for (int jj = 0; jj < 2; ++jj) {
        const int kk = c * 4 + half * 2 + jj;
        ga[r][c][jj] = Gws[row * K_DIM + kk];
      }
    }
  }

  // AtX / S / Y in C/D layout: VGPR v: lanes 0-15 row v, lanes 16-31 row v+8
  float atx[2][8], sreg[2][8], yreg[2][8];
  #pragma unroll
  for (int r = 0; r < 2; ++r) {
    #pragma unroll
    for (int v = 0; v < 8; ++v) {
      const int row = r * 16 + v + half * 8;
      atx[r][v] = AtX[row * N_DIM + col];
      sreg[r][v] = 0.f;
      yreg[r][v] = 0.f;
    }
  }

  float t = 1.0f;
  for (int it = 0; it < FISTA_ITERS; ++it) {
    float* buf = Ybuf[it & 1][wave];
    // Stage Y (C-layout regs) into plain [row][col] LDS tile
    #pragma unroll
    for (int r = 0; r < 2; ++r) {
      #pragma unroll
      for (int v = 0; v < 8; ++v) {
        const int row = r * 16 + v + half * 8;
        buf[row * 16 + colLoc] = yreg[r][v];
      }
    }
    __syncthreads();
    // Gather B-operand chunks (4x16 per k-chunk)
    v2f b[8];
    #pragma unroll
    for (int c = 0; c < 8; ++c) {
      #pragma unroll
      for (int jj = 0; jj < 2; ++jj) {
        const int kk = c * 4 + half * 2 + jj;
        b[c][jj] = buf[kk * 16 + colLoc];
      }
    }

    // GY = G @ Y : two 16x16 output row-blocks, K=32 as 8 chained k=4 WMMAs
    v8f acc0 = {};
    v8f acc1 = {};
    #pragma unroll
    for (int c = 0; c < 8; ++c) {
      acc0 = __builtin_amdgcn_wmma_f32_16x16x4_f32(false, ga[0][c], false,
                                                   b[c], (short)0, acc0,
                                                   false, false);
      acc1 = __builtin_amdgcn_wmma_f32_16x16x4_f32(false, ga[1][c], false,
                                                   b[c], (short)0, acc1,
                                                   false, false);
    }

    // FISTA scalar state: fast sqrt/rcp (momentum coefficient tolerates ulp
    // error); avoids the full IEEE divide expansion inside the hot loop.
    const float tn =
        0.5f * (1.0f + __builtin_amdgcn_sqrtf(fmaf(4.0f * t, t, 1.0f)));
    const float mom = (t - 1.0f) * __builtin_amdgcn_rcpf(tn);
    t = tn;

    // Projected gradient + Nesterov momentum, elementwise in C-layout
    #pragma unroll
    for (int r = 0; r < 2; ++r) {
      #pragma unroll
      for (int v = 0; v < 8; ++v) {
        const float gy = (r == 0) ? acc0[v] : acc1[v];
        float sn = yreg[r][v] - step * (gy - atx[r][v]);
        sn = fmaxf(sn, 0.0f);
        const float yn = sn + mom * (sn - sreg[r][v]);
        sreg[r][v] = sn;
        yreg[r][v] = yn;
      }
    }
  }

  #pragma unroll
  for (int r = 0; r < 2; ++r) {
    #pragma unroll
    for (int v = 0; v < 8; ++v) {
      const int row = r * 16 + v + half * 8;
      Sout[row * N_DIM + col] = sreg[r][v];
    }
  }
}

extern "C" void kernel_launch(void* const* d_in, const int* in_sizes, int n_in,
                              void* d_out, int out_size, void* d_ws,
                              size_t ws_size, hipStream_t stream) {
  const float* X = (const float*)d_in[0];  // (1024, 16384)
  const float* A = (const float*)d_in[1];  // (1024, 32)
  float* out = (float*)d_out;              // (32, 16384)
  float* ws = (float*)d_ws;
  float* G = ws;            // 1024 floats
  float* step = ws + 1024;  // 1 float
  float* AtX = ws + 2048;   // 32*16384 floats

  gram_step_kernel<<<1, 1024, 0, stream>>>(A, G, step);
  atx_kernel<<<N_DIM / 256, 256, 0, stream>>>(X, A, AtX);
  fista_kernel<<<N_DIM / 64, 128, 0, stream>>>(G, step, AtX, out);
}